// TFCriterion_55954833932854
// MI455X (gfx1250) — compile-verified
//
#include <hip/hip_runtime.h>
#include <hip/hip_bf16.h>

#define L_DIM 1024
#define B_DIM 32
#define F_DIM 24
#define S_DIM 12
#define N_TOT (L_DIM * B_DIM)
#define BAND  28   // exp(-d*d/8) == 0.0f (fp32) for |d| > 28 -> W is exactly banded

typedef __attribute__((ext_vector_type(2))) float v2f;
typedef __attribute__((ext_vector_type(8))) float v8f;

__device__ __forceinline__ v8f wmma_f32_16x16x4(v2f a, v2f b, v8f c) {
  // D = A(16x4) * B(4x16) + C, fp32 WMMA (VOP3P), wave32
  return __builtin_amdgcn_wmma_f32_16x16x4_f32(false, a, false, b, (short)0, c,
                                               false, false);
}

// ---------------------------------------------------------------------------
// Stage B: T_past/T_fut = W_past/W_fut (1024x1024, banded Gaussian, generated
// on the fly) x X (1024 x C). One wave = one 16x16 output tile for BOTH
// weight matrices (shared B fragment, two accumulators). K-loop restricted to
// the exact fp32-nonzero band: [m0-28, m0+44).
// ---------------------------------------------------------------------------
__global__ void __launch_bounds__(128)
gemm_msgs_kernel(const float* __restrict__ X, int C,
                 float* __restrict__ outP, float* __restrict__ outF)
{
  const int lane = threadIdx.x & 31;
  const int wv   = threadIdx.x >> 5;
  const int n0   = blockIdx.x * 64 + wv * 16;   // column tile
  const int m0   = blockIdx.y * 16;             // row tile
  const int half = lane >> 4;
  const int mr   = lane & 15;

  // A-fragment: lane (half,mr) supplies W[t, k0] and W[t, k0+1]
  const int   t    = m0 + mr;
  const float invP = 1.0f / (float)(t > 1 ? t : 1);
  const int   rem  = L_DIM - 1 - t;
  const float invF = 1.0f / (float)(rem > 1 ? rem : 1);

  int kmin = m0 - BAND;      if (kmin < 0) kmin = 0;          // multiple of 4
  int kmax = m0 + 16 + BAND; if (kmax > L_DIM) kmax = L_DIM;  // multiple of 4

  v8f accP = {};
  v8f accF = {};
  const float* __restrict__ Xn = X + n0 + mr;   // B frag column for this lane

  for (int kb = kmin; kb < kmax; kb += 4) {
    const int k0 = kb + 2 * half;
    // Gaussian message weights, generated in-register (2x v_exp_f32)
    float d0 = (float)(k0 - t);
    float d1 = d0 + 1.0f;
    float g0 = __expf(d0 * d0 * -0.125f);
    float g1 = __expf(d1 * d1 * -0.125f);
    v2f aP, aF, b;
    aP.x = (k0     < t) ? g0 * invP : 0.0f;
    aP.y = (k0 + 1 < t) ? g1 * invP : 0.0f;
    aF.x = (k0     > t) ? g0 * invF : 0.0f;
    aF.y = (k0 + 1 > t) ? g1 * invF : 0.0f;
    const float* xp = Xn + (size_t)k0 * C;      // B frag: X[k0,col], X[k0+1,col]
    b.x = xp[0];
    b.y = xp[C];
    accP = wmma_f32_16x16x4(aP, b, accP);
    accF = wmma_f32_16x16x4(aF, b, accF);
  }

  // D layout: VGPR r -> row m0 + r + 8*half, col n0 + (lane&15)
  const int col = n0 + mr;
#pragma unroll
  for (int r = 0; r < 8; ++r) {
    const size_t row = (size_t)(m0 + r + 8 * half);
    outP[row * C + col] = accP[r];
    outF[row * C + col] = accF[r];
  }
}

// ---------------------------------------------------------------------------
// wave32 reductions
// ---------------------------------------------------------------------------
__device__ __forceinline__ float wred_max(float v) {
#pragma unroll
  for (int o = 16; o > 0; o >>= 1) v = fmaxf(v, __shfl_xor(v, o, 32));
  return v;
}
__device__ __forceinline__ float wred_sum(float v) {
#pragma unroll
  for (int o = 16; o > 0; o >>= 1) v += __shfl_xor(v, o, 32);
  return v;
}

// softmax + NLL over lane-distributed logits (valid lanes hold v; invalid hold
// a huge negative). Writes per-lane probability, returns -log_softmax[label].
__device__ __forceinline__ float softmax_ce(float v, bool valid, int label,
                                            float* prob)
{
  float m  = wred_max(v);
  float e  = valid ? __expf(v - m) : 0.0f;
  float se = wred_sum(e);
  *prob = e / se;
  float vl = __shfl(v, label, 32);
  return -(vl - m - __logf(se));
}

// ---------------------------------------------------------------------------
// Async global->LDS staging (CDNA5 ASYNCcnt path).
// ---------------------------------------------------------------------------
__device__ __forceinline__ void async_b32(float* l, const float* g) {
  // vdst = LDS byte address (low 32 bits of the flat shared pointer),
  // vaddr = 64-bit global address, GV mode.
  asm volatile("global_load_async_to_lds_b32 %0, %1, off"
               :: "v"((unsigned)(unsigned long long)l), "v"(g)
               : "memory");
}

#define MAT_FLOATS 1584   // ff 576 | fs 288 | fs_t 288 | sf_t 288 | ss 144
#define OFF_FF  0
#define OFF_FS  576
#define OFF_FST 864
#define OFF_SFT 1152
#define OFF_SS  1440
#define MIX_WAVES 4       // waves per block
#define MIX_NPW   4       // n's per wave (double-buffered)
// async instructions per stage: 18 + 9 + 9 + 9 + 5 = 50 (0x32)

__device__ __forceinline__ void stage_n(const float* __restrict__ ff,
                                        const float* __restrict__ fs,
                                        const float* __restrict__ fs_t,
                                        const float* __restrict__ sf_t,
                                        const float* __restrict__ ss,
                                        int n, float* buf, int lane)
{
  // WAR guard: previous iteration's ds_loads from this buffer must retire
  // before the async DMA overwrites it (DScnt and ASYNCcnt are unordered).
  asm volatile("s_wait_dscnt 0x0" ::: "memory");
  const float* g;
  g = ff   + (size_t)n * 576;
  for (int i = lane; i < 576; i += 32) async_b32(buf + OFF_FF  + i, g + i);
  g = fs   + (size_t)n * 288;
  for (int i = lane; i < 288; i += 32) async_b32(buf + OFF_FS  + i, g + i);
  g = fs_t + (size_t)n * 288;
  for (int i = lane; i < 288; i += 32) async_b32(buf + OFF_FST + i, g + i);
  g = sf_t + (size_t)n * 288;
  for (int i = lane; i < 288; i += 32) async_b32(buf + OFF_SFT + i, g + i);
  g = ss   + (size_t)n * 144;
  for (int i = lane; i < 144; i += 32) async_b32(buf + OFF_SS  + i, g + i);
}

// ---------------------------------------------------------------------------
// Stage C: one wave per n (4 n's per wave, double-buffered async LDS staging).
// Single coalesced global pass per matrix block; all row/column access
// patterns served from LDS. Fused with all softmax/CE loss terms.
// ---------------------------------------------------------------------------
__global__ void __launch_bounds__(128)
mix_kernel(const float* __restrict__ f,    const float* __restrict__ s,
           const float* __restrict__ fs,   const float* __restrict__ ff,
           const float* __restrict__ ss,   const float* __restrict__ fs_t,
           const float* __restrict__ sf_t,
           const int* __restrict__ fl,     const int* __restrict__ sl,
           const int* __restrict__ yl,     const unsigned char* __restrict__ mask,
           const int* __restrict__ y2f,    const int* __restrict__ y2s,
           const float* __restrict__ f_mp, const float* __restrict__ f_mf,
           const float* __restrict__ s_mp, const float* __restrict__ s_mf,
           float* __restrict__ out_f,      float* __restrict__ out_s,
           float* __restrict__ acc)
{
  __shared__ float smem[MIX_WAVES * 2 * MAT_FLOATS];   // 50688 B

  const int lane  = threadIdx.x & 31;
  const int wv    = threadIdx.x >> 5;
  const int nbase = (blockIdx.x * MIX_WAVES + wv) * MIX_NPW;

  float* buf0 = smem + (size_t)(wv * 2) * MAT_FLOATS;
  float* buf1 = buf0 + MAT_FLOATS;

  const bool  inF = lane < F_DIM;
  const bool  inS = lane < S_DIM;
  const float NEG = -3.0e38f;

  // prime the pipeline
  stage_n(ff, fs, fs_t, sf_t, ss, nbase, buf0, lane);

  float loss_sum = 0.0f, mask_sum = 0.0f;

  for (int it = 0; it < MIX_NPW; ++it) {
    const int n  = nbase + it;
    float* buf   = (it & 1) ? buf1 : buf0;
    float* nxt   = (it & 1) ? buf0 : buf1;

    if (it + 1 < MIX_NPW) {
      stage_n(ff, fs, fs_t, sf_t, ss, n + 1, nxt, lane);
      // <=50 pending == only the just-issued batch -> current buffer landed
      asm volatile("s_wait_asynccnt 0x32" ::: "memory");
    } else {
      asm volatile("s_wait_asynccnt 0x0" ::: "memory");
    }

    const float* l_ff  = buf + OFF_FF;
    const float* l_fs  = buf + OFF_FS;
    const float* l_fst = buf + OFF_FST;
    const float* l_sft = buf + OFF_SFT;
    const float* l_ss  = buf + OFF_SS;

    float of  = inF ? f[(size_t)n * F_DIM + lane] : NEG;
    float os  = inS ? s[(size_t)n * S_DIM + lane] : NEG;
    float fmp = inF ? f_mp[(size_t)n * F_DIM + lane] : 0.0f;
    float fmf = inF ? f_mf[(size_t)n * F_DIM + lane] : 0.0f;
    float smp = inS ? s_mp[(size_t)n * S_DIM + lane] : 0.0f;
    float smf = inS ? s_mf[(size_t)n * S_DIM + lane] : 0.0f;

    // ---- next_f (lane = output channel g) ----
    float aF = 0.0f;
#pragma unroll
    for (int j = 0; j < F_DIM; ++j) {
      float wp = __shfl(fmp, j, 32);
      float wf = __shfl(fmf, j, 32);
      aF += wp * (inF ? l_ff[j * F_DIM + lane] : 0.0f);   // f_mp @ ff
      aF += (inF ? l_ff[lane * F_DIM + j] : 0.0f) * wf;   // ff @ f_mf
    }
#pragma unroll
    for (int j = 0; j < S_DIM; ++j) {
      float wp = __shfl(smp, j, 32);
      float wf = __shfl(smf, j, 32);
      float wo = __shfl(os,  j, 32);
      aF += wp * (inF ? l_sft[j * F_DIM + lane] : 0.0f);  // s_mp @ sf_t
      aF += (inF ? l_fst[lane * S_DIM + j] : 0.0f) * wf;  // fs_t @ s_mf
      aF += (inF ? l_fs [lane * S_DIM + j] : 0.0f) * wo;  // fs @ prev_s
    }
    float nf = (inF ? of : NEG) + 0.5f * aF;   // W_TEMPORAL == W_SPATIAL == 0.5

    // ---- next_s (lane = output channel t) ----
    float aS = 0.0f;
#pragma unroll
    for (int j = 0; j < S_DIM; ++j) {
      float wp = __shfl(smp, j, 32);
      float wf = __shfl(smf, j, 32);
      aS += wp * (inS ? l_ss[j * S_DIM + lane] : 0.0f);   // s_mp @ ss
      aS += (inS ? l_ss[lane * S_DIM + j] : 0.0f) * wf;   // ss @ s_mf
    }
#pragma unroll
    for (int j = 0; j < F_DIM; ++j) {
      float wp = __shfl(fmp, j, 32);
      float wf = __shfl(fmf, j, 32);
      float wo = __shfl(of,  j, 32);
      aS += wp * (inS ? l_fst[j * S_DIM + lane] : 0.0f);  // f_mp @ fs_t
      aS += (inS ? l_sft[lane * F_DIM + j] : 0.0f) * wf;  // sf_t @ f_mf
      aS += wo * (inS ? l_fs[j * S_DIM + lane] : 0.0f);   // prev_f @ fs
    }
    float ns_ = (inS ? os : NEG) + 0.5f * aS;

    // ---- losses + outputs ----
    const int lab_f = fl[n], lab_s = sl[n], lab_y = yl[n];
    float p_of, p_os, p_nf, p_ns;
    float nll_f0 = softmax_ce(of,  inF, lab_f, &p_of);
    float nll_s0 = softmax_ce(os,  inS, lab_s, &p_os);
    float nll_f1 = softmax_ce(nf,  inF, lab_f, &p_nf);
    float nll_s1 = softmax_ce(ns_, inS, lab_s, &p_ns);

    if (inF) out_f[(size_t)n * F_DIM + lane] = p_nf;
    if (inS) out_s[(size_t)n * S_DIM + lane] = p_ns;

    const int idxf = y2f[lab_y];
    const int idxs = y2s[lab_y];
    float strct = -(__shfl(p_of, idxf, 32) * __shfl(p_os, idxs, 32));

    float mk = mask[n] ? 1.0f : 0.0f;
    loss_sum += mk * (nll_f0 + nll_s0 + strct + nll_f1 + nll_s1);
    mask_sum += mk;
  }

  if (lane == 0) {
    atomicAdd(&acc[0], loss_sum);
    atomicAdd(&acc[1], mask_sum);
  }
}

__global__ void init_acc_kernel(float* acc) { acc[0] = 0.0f; acc[1] = 0.0f; }
__global__ void finalize_kernel(const float* acc, float* out_loss) {
  *out_loss = acc[0] / acc[1];
}

// ---------------------------------------------------------------------------
extern "C" void kernel_launch(void* const* d_in, const int* in_sizes, int n_in,
                              void* d_out, int out_size, void* d_ws, size_t ws_size,
                              hipStream_t stream) {
  (void)in_sizes; (void)n_in; (void)out_size; (void)ws_size;

  const float* f    = (const float*)d_in[0];
  const float* s    = (const float*)d_in[1];
  const float* fs   = (const float*)d_in[2];
  const float* ff   = (const float*)d_in[3];
  const float* ss   = (const float*)d_in[4];
  const float* fs_t = (const float*)d_in[5];
  const float* sf_t = (const float*)d_in[6];
  const int*   fl   = (const int*)d_in[7];
  const int*   sl   = (const int*)d_in[8];
  const int*   yl   = (const int*)d_in[9];
  const unsigned char* mask = (const unsigned char*)d_in[10];
  const int*   y2f  = (const int*)d_in[11];
  const int*   y2s  = (const int*)d_in[12];

  // workspace: [acc(16 floats pad)] [f_mp N*F] [f_mf N*F] [s_mp N*S] [s_mf N*S]
  float* acc  = (float*)d_ws;
  float* f_mp = acc + 16;
  float* f_mf = f_mp + (size_t)N_TOT * F_DIM;
  float* s_mp = f_mf + (size_t)N_TOT * F_DIM;
  float* s_mf = s_mp + (size_t)N_TOT * S_DIM;

  float* out_f    = (float*)d_out;
  float* out_s    = out_f + (size_t)N_TOT * F_DIM;
  float* out_loss = out_s + (size_t)N_TOT * S_DIM;

  init_acc_kernel<<<dim3(1), dim3(1), 0, stream>>>(acc);

  // message-passing GEMMs (banded, fp32 WMMA): f -> (f_mp, f_mf), s -> (s_mp, s_mf)
  {
    dim3 g((B_DIM * F_DIM) / 64, L_DIM / 16);   // 12 x 64 tiles
    gemm_msgs_kernel<<<g, dim3(128), 0, stream>>>(f, B_DIM * F_DIM, f_mp, f_mf);
  }
  {
    dim3 g((B_DIM * S_DIM) / 64, L_DIM / 16);   // 6 x 64 tiles
    gemm_msgs_kernel<<<g, dim3(128), 0, stream>>>(s, B_DIM * S_DIM, s_mp, s_mf);
  }

  // per-n mixing + softmax + all CE/struct loss terms
  mix_kernel<<<dim3(N_TOT / (MIX_WAVES * MIX_NPW)), dim3(32 * MIX_WAVES), 0, stream>>>(
      f, s, fs, ff, ss, fs_t, sf_t, fl, sl, yl, mask, y2f, y2s,
      f_mp, f_mf, s_mp, s_mf, out_f, out_s, acc);

  finalize_kernel<<<dim3(1), dim3(1), 0, stream>>>(acc, out_loss);
}